// VectorQuantizer_27264452395730
// MI455X (gfx1250) — compile-verified
//
#include <hip/hip_runtime.h>
#include <hip/hip_bf16.h>
#include <math.h>

// ---------------------------------------------------------------------------
// Sinkhorn Vector Quantizer for MI455X (gfx1250).
//   N = 8192 latents, K = 4096 codes, D = 512, fp32.
// Phase 1: d[k][n] = ||e_k||^2 + ||x_n||^2 - 2 e_k.x_n via V_WMMA_F32_16X16X4_F32,
//          tiles staged into LDS by the Tensor Data Mover (tensor_load_to_lds,
//          double-buffered, TDM hardware padding makes the 34-dword LDS rows).
// Phase 2: Sinkhorn in log space reduced to offset updates:
//          u[k] = -LSE_n(lq0+v[n]) - logK ; v[n] = -LSE_k(lq0+u[k]) - logB
//          with lq0(k,n) = (mid - d(k,n)) / (amp*eps).  d (134 MB) stays
//          resident in the 192 MB L2; each pass is a streaming logsumexp.
// Phase 3: argmax_k(lq0 + u[k]), gather, loss = 1.25 * mean((x_q - x)^2).
// All reductions use fixed-order two-stage combines (deterministic).
// ---------------------------------------------------------------------------

#define N_LAT 8192
#define K_CODE 4096
#define DIM 512
#define XQ_SIZE (N_LAT * DIM)         // 4194304
#define LOG_K 8.317766166719343f      // log(4096)
#define LOG_B 9.010913347279288f      // log(8192)
#define SK_EPS 0.003f
#define SK_ITERS 100

typedef float v2f __attribute__((ext_vector_type(2)));
typedef float v8f __attribute__((ext_vector_type(8)));
typedef unsigned int u32x4 __attribute__((ext_vector_type(4)));
typedef int i32x4 __attribute__((ext_vector_type(4)));
typedef int i32x8 __attribute__((ext_vector_type(8)));

__device__ __forceinline__ unsigned int fenc(float f) {
  unsigned int b = __float_as_uint(f);
  return (b & 0x80000000u) ? ~b : (b | 0x80000000u);
}
__device__ __forceinline__ float fdec(unsigned int u) {
  unsigned int b = (u & 0x80000000u) ? (u & 0x7FFFFFFFu) : ~u;
  return __uint_as_float(b);
}

// ---------------------------------------------------------------------------
// TDM 2D tile load: 128 rows x 32 f32, row stride 512 f32, into LDS with
// hardware padding (pad_interval=32 dwords -> +2 dwords) => 34-dword rows.
// D# layout per CDNA5 ISA ch.8 (groups 0/1; groups 2/3 zero for 2D).
// This toolchain exposes the 6-arg builtin: (g0, g1, g2, g3, g4, cpol).
// ---------------------------------------------------------------------------
__device__ __forceinline__ void tdm_load_tile(const float* gptr,
                                              unsigned lds_off,
                                              unsigned tensor_rows) {
  unsigned long long ga = (unsigned long long)gptr;
  u32x4 g0;
  g0[0] = 1u;                                   // count=1 valid descriptor
  g0[1] = lds_off;                              // lds_addr (bytes)
  g0[2] = (unsigned)(ga & 0xFFFFFFFFu);         // global_addr[31:0]
  g0[3] = (unsigned)((ga >> 32) & 0x01FFFFFFu)  // global_addr[56:32]
          | (2u << 30);                         // type=2 (image)
  i32x8 g1;
  g1[0] = (int)((2u << 16)      // data_size = 4B
                | (1u << 20)    // pad_enable
                | (4u << 22)    // pad_interval: 32 dwords
                | (1u << 25));  // pad_amount: 2 dwords
  g1[1] = (int)(512u << 16);                    // tensor_dim0[15:0] = 512
  g1[2] = (int)((tensor_rows & 0xFFFFu) << 16); // tensor_dim1[15:0]
  g1[3] = (int)(((tensor_rows >> 16) & 0xFFFFu) // tensor_dim1[31:16]
                | (32u << 16));                 // tile_dim0 = 32
  g1[4] = 128;                                  // tile_dim1 = 128
  g1[5] = 512;                                  // tensor_dim0_stride[31:0]
  g1[6] = 0;
  g1[7] = 0;
  i32x4 z4 = {0, 0, 0, 0};
  i32x8 z8 = {0, 0, 0, 0, 0, 0, 0, 0};
  __builtin_amdgcn_tensor_load_to_lds(g0, g1, z4, z4, z8, 0);
}

// --------------------------------------------------------------------------
// init: v[n] = 0, seed global min/max encodings
// --------------------------------------------------------------------------
__global__ __launch_bounds__(256) void vq_init(float* __restrict__ v,
                                               unsigned int* __restrict__ sc) {
  int i = blockIdx.x * 256 + threadIdx.x;
  if (i < N_LAT) v[i] = 0.0f;
  if (i == 0) { sc[0] = 0u; sc[1] = 0xFFFFFFFFu; }
}

// --------------------------------------------------------------------------
// row squared norms for x (blocks 0..8191) and embedding (blocks 8192..12287)
// --------------------------------------------------------------------------
__global__ __launch_bounds__(256) void vq_norms(const float* __restrict__ x,
                                                const float* __restrict__ e,
                                                float* __restrict__ xsq,
                                                float* __restrict__ esq) {
  __shared__ float red[256];
  int r = blockIdx.x, t = threadIdx.x;
  const float* p = (r < N_LAT) ? (x + (size_t)r * DIM)
                               : (e + (size_t)(r - N_LAT) * DIM);
  float a = p[t], b = p[t + 256];
  red[t] = a * a + b * b;
  __syncthreads();
  for (int off = 128; off > 0; off >>= 1) {
    if (t < off) red[t] += red[t + off];
    __syncthreads();
  }
  if (t == 0) {
    if (r < N_LAT) xsq[r] = red[0];
    else esq[r - N_LAT] = red[0];
  }
}

// --------------------------------------------------------------------------
// GEMM + distance epilogue + global min/max, fp32 WMMA 16x16x4.
// Output dt[k][n], k-major (K=4096 rows, N=8192 cols).
// Block tile 128(M=k) x 128(N=n); 8 waves as 2x4; wave tile 64x32 = 4x2 frags.
// Both tiles LDS-resident row-major [row][d] (pad 34) so every A/B fragment
// is one contiguous 8-byte LDS read. Staging via TDM, double-buffered.
// --------------------------------------------------------------------------
__global__ __launch_bounds__(256) void vq_gemm_dist(
    const float* __restrict__ x, const float* __restrict__ e,
    const float* __restrict__ xsq, const float* __restrict__ esq,
    float* __restrict__ dt, unsigned int* __restrict__ sc) {
  __shared__ float As[2][128][34];   // e tile  [m][d]
  __shared__ float Xs[2][128][34];   // x tile  [n][d]
  __shared__ float redmin[256], redmax[256];

  const int tid = threadIdx.x;
  const int bn = blockIdx.x;      // n tile
  const int bm = blockIdx.y;      // k tile
  const int w = tid >> 5, lane = tid & 31;
  const int wm = w >> 2, wn = w & 3;      // 2 x 4 wave grid
  const int l16 = lane & 15, lhi = lane >> 4;
  const bool w0 = (w == 0);

  const float* eBase = e + (size_t)(bm * 128) * DIM;
  const float* xBase = x + (size_t)(bn * 128) * DIM;

  v8f acc[4][2] = {};

  if (w0) {  // TDM: preload chunk 0 (EXEC-independent, per-wave TENSORcnt)
    tdm_load_tile(eBase, (unsigned)(unsigned long long)&As[0][0][0], K_CODE);
    tdm_load_tile(xBase, (unsigned)(unsigned long long)&Xs[0][0][0], N_LAT);
  }

  for (int c = 0; c < DIM / 32; ++c) {
    if (w0) __builtin_amdgcn_s_wait_tensorcnt((short)0);
    __syncthreads();               // buffer (c&1) ready for everyone
    const int cur = c & 1;
    if (w0 && c + 1 < DIM / 32) {  // overlap next DMA with compute
      const int nxt = (c + 1) & 1;
      tdm_load_tile(eBase + (c + 1) * 32,
                    (unsigned)(unsigned long long)&As[nxt][0][0], K_CODE);
      tdm_load_tile(xBase + (c + 1) * 32,
                    (unsigned)(unsigned long long)&Xs[nxt][0][0], N_LAT);
    }
#pragma unroll
    for (int ds = 0; ds < 8; ++ds) {
      // ISA 16x4 f32 A/B layout: lanes0-15 hold K={0,1}, lanes16-31 K={2,3}
      int kb = ds * 4 + lhi * 2;
      v2f a[4], b[2];
#pragma unroll
      for (int mt = 0; mt < 4; ++mt) {
        int row = wm * 64 + mt * 16 + l16;
        a[mt] = *(const v2f*)&As[cur][row][kb];
      }
#pragma unroll
      for (int nt = 0; nt < 2; ++nt) {
        int col = wn * 32 + nt * 16 + l16;
        b[nt] = *(const v2f*)&Xs[cur][col][kb];
      }
#pragma unroll
      for (int mt = 0; mt < 4; ++mt)
#pragma unroll
        for (int nt = 0; nt < 2; ++nt)
          acc[mt][nt] = __builtin_amdgcn_wmma_f32_16x16x4_f32(
              false, a[mt], false, b[nt], (short)0, acc[mt][nt], false, false);
    }
    __syncthreads();               // all reads of buffer (c&1) done
  }

  // epilogue: d = ||e||^2 + ||x||^2 - 2 dot ; write + track min/max
  float lmin = INFINITY, lmax = -INFINITY;
#pragma unroll
  for (int mt = 0; mt < 4; ++mt) {
#pragma unroll
    for (int nt = 0; nt < 2; ++nt) {
      int gn = bn * 128 + wn * 32 + nt * 16 + l16;
#pragma unroll
      for (int g = 0; g < 8; ++g) {
        int gk = bm * 128 + wm * 64 + mt * 16 + lhi * 8 + g;  // C: M=g(+8)
        float dv = esq[gk] + xsq[gn] - 2.0f * acc[mt][nt][g];
        dt[(size_t)gk * N_LAT + gn] = dv;
        lmin = fminf(lmin, dv);
        lmax = fmaxf(lmax, dv);
      }
    }
  }
  redmin[tid] = lmin; redmax[tid] = lmax;
  __syncthreads();
  for (int off = 128; off > 0; off >>= 1) {
    if (tid < off) {
      redmin[tid] = fminf(redmin[tid], redmin[tid + off]);
      redmax[tid] = fmaxf(redmax[tid], redmax[tid + off]);
    }
    __syncthreads();
  }
  if (tid == 0) {
    atomicMax(&sc[0], fenc(redmax[0]));
    atomicMin(&sc[1], fenc(redmin[0]));
  }
}

// --------------------------------------------------------------------------
// finalize scale: mid = (mx+mn)/2, amp = mx-mid+1e-5, s = 1/(amp*eps)
// --------------------------------------------------------------------------
__global__ void vq_finalize(unsigned int* __restrict__ sc) {
  if (threadIdx.x == 0 && blockIdx.x == 0) {
    float mx = fdec(sc[0]), mn = fdec(sc[1]);
    float mid = 0.5f * (mx + mn);
    float amp = mx - mid + 1e-5f;
    float* f = (float*)sc;
    f[2] = mid;
    f[3] = 1.0f / (amp * SK_EPS);
  }
}

// --------------------------------------------------------------------------
// u[k] = -LSE_n((mid - d[k][n])*s + v[n]) - logK ; one block per k row
// --------------------------------------------------------------------------
__global__ __launch_bounds__(256) void vq_sink_u(
    const float* __restrict__ dt, const float* __restrict__ v,
    float* __restrict__ u, const unsigned int* __restrict__ sc) {
  __shared__ float vv[N_LAT];
  __shared__ float rm[256], ra[256];
  int k = blockIdx.x, t = threadIdx.x;
  for (int i = t; i < N_LAT; i += 256) vv[i] = v[i];
  __syncthreads();
  const float* f = (const float*)sc;
  float mid = f[2], s = f[3];
  const float* row = dt + (size_t)k * N_LAT;
  float m = -INFINITY, a = 0.0f;
#pragma unroll 4
  for (int j = 0; j < N_LAT / 256; ++j) {
    int n = t + j * 256;
    float val = (mid - row[n]) * s + vv[n];
    if (val > m) { a = a * __expf(m - val) + 1.0f; m = val; }
    else a += __expf(val - m);
  }
  rm[t] = m; ra[t] = a;
  __syncthreads();
  for (int off = 128; off > 0; off >>= 1) {
    if (t < off) {
      float m2 = rm[t + off], a2 = ra[t + off];
      float M = fmaxf(rm[t], m2);
      ra[t] = ra[t] * __expf(rm[t] - M) + a2 * __expf(m2 - M);
      rm[t] = M;
    }
    __syncthreads();
  }
  if (t == 0) u[k] = -(rm[0] + __logf(ra[0])) - LOG_K;
}

// --------------------------------------------------------------------------
// v[n] = -LSE_k((mid - d[k][n])*s + u[k]) - logB ; 32 cols/block, 8 k-groups
// --------------------------------------------------------------------------
__global__ __launch_bounds__(256) void vq_sink_v(
    const float* __restrict__ dt, const float* __restrict__ u,
    float* __restrict__ v, const unsigned int* __restrict__ sc) {
  __shared__ float uu[K_CODE];
  __shared__ float rm[8][32], ra[8][32];
  int t = threadIdx.x;
  int cl = t & 31, kg = t >> 5;
  int col = blockIdx.x * 32 + cl;
  for (int i = t; i < K_CODE; i += 256) uu[i] = u[i];
  __syncthreads();
  const float* f = (const float*)sc;
  float mid = f[2], s = f[3];
  float m = -INFINITY, a = 0.0f;
  int k0 = kg * (K_CODE / 8);
  for (int k = k0; k < k0 + K_CODE / 8; ++k) {
    float val = (mid - dt[(size_t)k * N_LAT + col]) * s + uu[k];
    if (val > m) { a = a * __expf(m - val) + 1.0f; m = val; }
    else a += __expf(val - m);
  }
  rm[kg][cl] = m; ra[kg][cl] = a;
  __syncthreads();
  if (t < 32) {
    float M = rm[0][t], A = ra[0][t];
#pragma unroll
    for (int g = 1; g < 8; ++g) {
      float m2 = rm[g][t], a2 = ra[g][t];
      float MM = fmaxf(M, m2);
      A = A * __expf(M - MM) + a2 * __expf(m2 - MM);
      M = MM;
    }
    v[blockIdx.x * 32 + t] = -(M + __logf(A)) - LOG_B;
  }
}

// --------------------------------------------------------------------------
// argmax_k((mid - d[k][n])*s + u[k]) ; first-max semantics (ascending k)
// --------------------------------------------------------------------------
__global__ __launch_bounds__(256) void vq_argmax(
    const float* __restrict__ dt, const float* __restrict__ u,
    const unsigned int* __restrict__ sc, int* __restrict__ idx,
    float* __restrict__ out_idx) {
  __shared__ float uu[K_CODE];
  __shared__ float rb[8][32];
  __shared__ int rk[8][32];
  int t = threadIdx.x;
  int cl = t & 31, kg = t >> 5;
  int col = blockIdx.x * 32 + cl;
  for (int i = t; i < K_CODE; i += 256) uu[i] = u[i];
  __syncthreads();
  const float* f = (const float*)sc;
  float mid = f[2], s = f[3];
  float best = -INFINITY;
  int bk = 0;
  int k0 = kg * (K_CODE / 8);
  for (int k = k0; k < k0 + K_CODE / 8; ++k) {
    float val = (mid - dt[(size_t)k * N_LAT + col]) * s + uu[k];
    if (val > best) { best = val; bk = k; }
  }
  rb[kg][cl] = best; rk[kg][cl] = bk;
  __syncthreads();
  if (t < 32) {
    float B = rb[0][t];
    int K0 = rk[0][t];
#pragma unroll
    for (int g = 1; g < 8; ++g) {
      if (rb[g][t] > B) { B = rb[g][t]; K0 = rk[g][t]; }
    }
    int n = blockIdx.x * 32 + t;
    idx[n] = K0;
    out_idx[n] = (float)K0;
  }
}

// --------------------------------------------------------------------------
// gather x_q = embedding[idx], per-row loss partial sums
// --------------------------------------------------------------------------
__global__ __launch_bounds__(256) void vq_gather(
    const float* __restrict__ x, const float* __restrict__ e,
    const int* __restrict__ idx, float* __restrict__ xq,
    float* __restrict__ lp) {
  __shared__ float red[256];
  int n = blockIdx.x, t = threadIdx.x;
  int code = idx[n];
  const float* er = e + (size_t)code * DIM;
  const float* xr = x + (size_t)n * DIM;
  float* o = xq + (size_t)n * DIM;
  float ps = 0.0f;
#pragma unroll
  for (int j = 0; j < 2; ++j) {
    int d = t + j * 256;
    float q = er[d];
    float diff = q - xr[d];
    o[d] = q;                 // x_q_st value == x_q
    ps += diff * diff;
  }
  red[t] = ps;
  __syncthreads();
  for (int off = 128; off > 0; off >>= 1) {
    if (t < off) red[t] += red[t + off];
    __syncthreads();
  }
  if (t == 0) lp[n] = red[0];
}

// --------------------------------------------------------------------------
// final loss = (1 + beta) * mean = 1.25 * sum / (N*D) ; deterministic order
// --------------------------------------------------------------------------
__global__ __launch_bounds__(256) void vq_loss(const float* __restrict__ lp,
                                               float* __restrict__ loss) {
  __shared__ float red[256];
  int t = threadIdx.x;
  float s = 0.0f;
#pragma unroll
  for (int j = 0; j < N_LAT / 256; ++j) s += lp[t + j * 256];
  red[t] = s;
  __syncthreads();
  for (int off = 128; off > 0; off >>= 1) {
    if (t < off) red[t] += red[t + off];
    __syncthreads();
  }
  if (t == 0) *loss = 1.25f * red[0] / (float)XQ_SIZE;
}

// ---------------------------------------------------------------------------
extern "C" void kernel_launch(void* const* d_in, const int* in_sizes, int n_in,
                              void* d_out, int out_size, void* d_ws,
                              size_t ws_size, hipStream_t stream) {
  const float* x = (const float*)d_in[0];        // [8192, 512]
  const float* e = (const float*)d_in[1];        // [4096, 512]
  float* out = (float*)d_out;

  char* ws = (char*)d_ws;
  const size_t SZ_DT = (size_t)K_CODE * N_LAT * 4;   // 134 MB
  float* dt  = (float*)(ws);
  float* xsq = (float*)(ws + SZ_DT);
  float* esq = (float*)(ws + SZ_DT + 32768);
  float* u   = (float*)(ws + SZ_DT + 32768 + 16384);
  float* v   = (float*)(ws + SZ_DT + 32768 + 16384 + 16384);
  int*  idx  = (int*)  (ws + SZ_DT + 32768 + 16384 + 16384 + 32768);
  float* lp  = (float*)(ws + SZ_DT + 32768 + 16384 + 16384 + 32768 + 32768);
  unsigned int* sc =
      (unsigned int*)(ws + SZ_DT + 32768 + 16384 + 16384 + 32768 + 32768 + 32768);

  float* out_xq   = out;                 // 4194304 floats
  float* out_loss = out + XQ_SIZE;       // 1 float
  float* out_idx  = out + XQ_SIZE + 1;   // 8192 floats

  vq_init<<<N_LAT / 256, 256, 0, stream>>>(v, sc);
  vq_norms<<<N_LAT + K_CODE, 256, 0, stream>>>(x, e, xsq, esq);
  vq_gemm_dist<<<dim3(N_LAT / 128, K_CODE / 128), 256, 0, stream>>>(
      x, e, xsq, esq, dt, sc);
  vq_finalize<<<1, 64, 0, stream>>>(sc);

  for (int it = 0; it < SK_ITERS; ++it) {
    vq_sink_u<<<K_CODE, 256, 0, stream>>>(dt, v, u, sc);
    vq_sink_v<<<N_LAT / 32, 256, 0, stream>>>(dt, u, v, sc);
  }

  vq_argmax<<<N_LAT / 32, 256, 0, stream>>>(dt, u, sc, idx, out_idx);
  vq_gather<<<N_LAT, 256, 0, stream>>>(x, e, idx, out_xq, lp);
  vq_loss<<<1, 256, 0, stream>>>(lp, out_loss);
}